// DGFLayer_17987323035753
// MI455X (gfx1250) — compile-verified
//
#include <hip/hip_runtime.h>
#include <math.h>

// Problem constants (match reference)
#define B_ 8
#define N_ 1024
#define D_ 256
#define H_ 8

typedef __attribute__((ext_vector_type(2))) float v2f;
typedef __attribute__((ext_vector_type(4))) float v4f;
typedef __attribute__((ext_vector_type(8))) float v8f;

// Exact fp32 WMMA: D(16x16,f32) = A(16x4,f32) * B(4x16,f32) + C
// args: (neg_a, A, neg_b, B, c_mod, C, reuse_a, reuse_b)
__device__ __forceinline__ v8f wmma4(v2f a, v2f b, v8f c) {
  return __builtin_amdgcn_wmma_f32_16x16x4_f32(false, a, false, b, (short)0, c,
                                               false, false);
}

// ---------------------------------------------------------------------------
// Prep kernel: sq[b,n] = sum_d x^2 ; coeff[h] = -1/(2*sigma_h^2+1e-6) ; Wt = W^T
// ws layout (floats): sq[0..8192) | coeff[8192..8200) | Wt[8448..8448+65536)
// ---------------------------------------------------------------------------
__global__ __launch_bounds__(256) void dgf_prep(const float* __restrict__ x,
                                                const float* __restrict__ ls,
                                                const float* __restrict__ W,
                                                float* __restrict__ ws) {
  float* sq    = ws;
  float* coeff = ws + 8192;
  float* Wt    = ws + 8448;
  const int blk = blockIdx.x, t = threadIdx.x;
  if (blk < 32) {                       // row sum-of-squares, 8192 rows
    const int row = blk * 256 + t;
    const v4f* p = (const v4f*)(x + (size_t)row * D_);
    float s = 0.0f;
    #pragma unroll
    for (int i = 0; i < 64; ++i) {
      v4f v = p[i];
      s += v.x * v.x + v.y * v.y + v.z * v.z + v.w * v.w;
    }
    sq[row] = s;
  } else if (blk < 288) {               // Wt[k*D + d] = W[d*D + k]
    const int idx = (blk - 32) * 256 + t;
    const int k = idx >> 8, d = idx & 255;
    Wt[idx] = W[d * D_ + k];
  } else {
    if (t < H_) {
      float s = __expf(ls[t]);
      coeff[t] = -1.0f / (2.0f * s * s + 1e-6f);
    }
  }
}

// ---------------------------------------------------------------------------
// Fused kernel: per workgroup = 16 output rows of one batch. 8 waves (wave32).
//   phase B: flash-style loop over m-tiles: G-tile (WMMA) -> exp adjacency
//            -> Y += adj * X (WMMA); adjacency never leaves the WGP.
//   phase C: z = Y * W^T + b (WMMA), ELU, residual
//   phase D: LayerNorm epilogue
// A-fragments are sourced from LDS (keeps VGPRs < 256: no MSB churn, higher
// occupancy); B-fragments stream from L2 with 1-step software double-buffer.
// ---------------------------------------------------------------------------
__global__ __launch_bounds__(256) void dgf_main(const float* __restrict__ x,
                                                const float* __restrict__ b_proj,
                                                const float* __restrict__ gamma,
                                                const float* __restrict__ beta,
                                                const float* __restrict__ ws,
                                                float* __restrict__ out) {
  __shared__ float xrow[16 * 256];   // this tile's rows of x (A-source + residual)
  __shared__ float Ybuf[16 * 256];   // cross-wave Y reduction
  __shared__ float Ebuf[16 * 256];   // elu(z)+x, pre-LN
  __shared__ float adjst[8 * 256];   // per-wave adj tile (C/D -> A layout bounce)
  __shared__ float psum[256], psum2[256];
  __shared__ float meanv[16], rstdv[16];

  const float* sq    = ws;
  const float* coeff = ws + 8192;
  const float* Wt    = ws + 8448;

  const int t    = threadIdx.x;
  const int wave = t >> 5;           // wave32
  const int lane = t & 31;
  const int l15  = lane & 15;
  const int hi   = lane >> 4;        // lane half selects K pair / row half
  const int kofs = hi << 1;

  const int b        = blockIdx.x >> 6;          // 64 row-tiles per batch
  const int row_base = (blockIdx.x & 63) << 4;
  const float* xb = x + (size_t)b * N_ * D_;

  // stage x row-tile (16x256, contiguous in memory) + zero Y
  #pragma unroll
  for (int i = 0; i < 16; ++i) {
    xrow[i * 256 + t] = xb[(size_t)row_base * D_ + i * 256 + t];
    Ybuf[i * 256 + t] = 0.0f;
  }
  __syncthreads();

  // uniform per-head coefficients and this tile's row norms
  float c[H_];
  #pragma unroll
  for (int h = 0; h < H_; ++h) c[h] = coeff[h];
  float sqr[8];
  #pragma unroll
  for (int v = 0; v < 8; ++v) sqr[v] = sq[b * N_ + row_base + v + 8 * hi];

  v8f yacc[16];                       // Y[16 rows x 256 d] partial, per wave
  #pragma unroll
  for (int d = 0; d < 16; ++d) yacc[d] = (v8f)(0.0f);

  float* ast = adjst + wave * 256;    // private per-wave staging slot
  const float* xrA = &xrow[l15 * 256 + kofs];   // A-fragment base in LDS

  for (int mi = wave; mi < 64; mi += 8) {       // waves split the m dimension
    const int m_base = mi << 4;

    // warm WGP$ for the next m-chunk of this wave (global_prefetch_b8)
    if (mi + 8 < 64)
      __builtin_prefetch(xb + (size_t)(m_base + 128 + l15) * D_ + kofs, 0, 3);

    // ---- Gram tile G = Xrow * Xm^T (B-frag == A-frag of the m rows) ----
    const float* pB = xb + (size_t)(m_base + l15) * D_ + kofs;
    v2f cb0 = *(const v2f*)(pB);
    v2f cb1 = *(const v2f*)(pB + 4);
    v8f g0 = (v8f)(0.0f), g1 = (v8f)(0.0f);    // 2 accums break D->A chain
    #pragma unroll
    for (int ks = 0; ks < 64; ks += 2) {
      v2f nb0 = cb0, nb1 = cb1;                // 1-step double buffer
      if (ks + 2 < 64) {
        nb0 = *(const v2f*)(pB + 4 * (ks + 2));
        nb1 = *(const v2f*)(pB + 4 * (ks + 3));
      }
      v2f a0 = *(const v2f*)(xrA + 4 * ks);
      v2f a1 = *(const v2f*)(xrA + 4 * (ks + 1));
      g0 = wmma4(a0, cb0, g0);
      g1 = wmma4(a1, cb1, g1);
      cb0 = nb0; cb1 = nb1;
    }
    v8f g = g0 + g1;

    // ---- Gaussian kernel adjacency, mean over heads ----
    const float sqc = sq[b * N_ + m_base + l15];
    #pragma unroll
    for (int v = 0; v < 8; ++v) {
      float dist = fmaxf(sqr[v] + sqc - 2.0f * g[v], 0.0f);
      float a = 0.0f;
      #pragma unroll
      for (int h = 0; h < H_; ++h) a += __expf(dist * c[h]);
      ast[(v + 8 * hi) * 16 + l15] = a * 0.125f;  // C/D layout -> row-major LDS
    }

    // ---- Y += adj_tile(16x16) * Xm(16x256) ----
    #pragma unroll
    for (int ks2 = 0; ks2 < 4; ++ks2) {
      v2f aa = *(const v2f*)&ast[l15 * 16 + 4 * ks2 + kofs];   // A layout
      const float* pXm = xb + (size_t)(m_base + 4 * ks2 + kofs) * D_ + l15;
      v2f cbb;                                  // B feed, double-buffered
      cbb.x = pXm[0];
      cbb.y = pXm[D_];
      #pragma unroll
      for (int dt = 0; dt < 16; ++dt) {
        v2f nbb = cbb;
        if (dt + 1 < 16) {
          nbb.x = pXm[(dt + 1) * 16];           // coalesced across lanes
          nbb.y = pXm[D_ + (dt + 1) * 16];
        }
        yacc[dt] = wmma4(aa, cbb, yacc[dt]);
        cbb = nbb;
      }
    }
  }

  // cross-wave reduction of Y into LDS (ds_add_f32)
  #pragma unroll
  for (int dt = 0; dt < 16; ++dt)
    #pragma unroll
    for (int v = 0; v < 8; ++v)
      atomicAdd(&Ybuf[(v + 8 * hi) * 256 + dt * 16 + l15], yacc[dt][v]);
  __syncthreads();

  // ---- projection z = Y * W^T + b, ELU, residual (2 d-tiles per wave) ----
  #pragma unroll
  for (int dd = 0; dd < 2; ++dd) {
    const int dbase = (wave * 2 + dd) * 16;
    const float* pW = Wt + (size_t)kofs * D_ + dbase + l15;
    v8f z0 = (v8f)(0.0f), z1 = (v8f)(0.0f);
    #pragma unroll
    for (int ks = 0; ks < 64; ks += 2) {
      v2f ya0 = *(const v2f*)&Ybuf[l15 * 256 + 4 * ks + kofs];
      v2f ya1 = *(const v2f*)&Ybuf[l15 * 256 + 4 * ks + 4 + kofs];
      v2f wb0, wb1;
      wb0.x = pW[(size_t)(4 * ks) * D_];
      wb0.y = pW[(size_t)(4 * ks + 1) * D_];
      wb1.x = pW[(size_t)(4 * ks + 4) * D_];
      wb1.y = pW[(size_t)(4 * ks + 5) * D_];
      z0 = wmma4(ya0, wb0, z0);
      z1 = wmma4(ya1, wb1, z1);
    }
    v8f z = z0 + z1;
    #pragma unroll
    for (int v = 0; v < 8; ++v) {
      const int M = v + 8 * hi;
      const int d = dbase + l15;
      float zz = z[v] + b_proj[d];
      float el = (zz > 0.0f) ? zz : (__expf(zz) - 1.0f);   // ELU
      Ebuf[M * 256 + d] = el + xrow[M * 256 + d];          // residual
    }
  }
  __syncthreads();

  // ---- LayerNorm over D=256 ----
  {
    const int r = t >> 4, c0 = t & 15;
    float s = 0.0f, s2 = 0.0f;
    #pragma unroll
    for (int j = 0; j < 16; ++j) {
      float v = Ebuf[r * 256 + c0 + 16 * j];
      s += v; s2 += v * v;
    }
    psum[t] = s; psum2[t] = s2;
  }
  __syncthreads();
  if (t < 16) {
    float s = 0.0f, s2 = 0.0f;
    #pragma unroll
    for (int j = 0; j < 16; ++j) { s += psum[t * 16 + j]; s2 += psum2[t * 16 + j]; }
    const float mean = s * (1.0f / 256.0f);
    const float var  = s2 * (1.0f / 256.0f) - mean * mean;
    meanv[t] = mean;
    rstdv[t] = rsqrtf(var + 1e-5f);
  }
  __syncthreads();
  {
    const int r = t >> 4, c0 = t & 15;
    const float mean = meanv[r], rstd = rstdv[r];
    float* og = out + ((size_t)b * N_ + row_base + r) * D_;
    #pragma unroll
    for (int j = 0; j < 16; ++j) {
      const int d = c0 + 16 * j;
      og[d] = gamma[d] * (Ebuf[r * 256 + d] - mean) * rstd + beta[d];
    }
  }
}

// ---------------------------------------------------------------------------
extern "C" void kernel_launch(void* const* d_in, const int* in_sizes, int n_in,
                              void* d_out, int out_size, void* d_ws,
                              size_t ws_size, hipStream_t stream) {
  const float* x  = (const float*)d_in[0];   // [B,N,D]
  const float* ls = (const float*)d_in[1];   // [H]
  const float* W  = (const float*)d_in[2];   // [D,D]
  const float* bp = (const float*)d_in[3];   // [D]
  const float* gm = (const float*)d_in[4];   // [D]
  const float* bt = (const float*)d_in[5];   // [D]
  float* ws  = (float*)d_ws;                 // needs ~290 KB
  float* out = (float*)d_out;                // [B,N,D]

  dgf_prep<<<289, 256, 0, stream>>>(x, ls, W, ws);
  dgf_main<<<B_ * (N_ / 16), 256, 0, stream>>>(x, bp, gm, bt, ws, out);
}